// HybridDetectionModel_48344151884162
// MI455X (gfx1250) — compile-verified
//
#include <hip/hip_runtime.h>
#include <math.h>
#include <stdint.h>

#define NCLS 80
#define ROWS 8                       // rows per tile per wave
#define WAVES 8                      // waves per 256-thread block
#define TILE_BYTES (ROWS * NCLS * 4) // 2560 B per tile
#define COPIES (TILE_BYTES / 512)    // 5 x b128 per wave (32 lanes * 16B = 512B each)

__device__ __forceinline__ float warp_sum_f32(float v) {
#pragma unroll
    for (int off = 16; off > 0; off >>= 1) v += __shfl_xor(v, off, 32);
    return v;
}

// ---------------------------------------------------------------------------
// Fused loss kernel:
//   Part 1: focal loss — wave-per-row-tile, async global->LDS double-buffered
//           staging of the 320 MB logits stream (ASYNCcnt pipeline).
//   Part 2: GIoU + centerness — thread-per-row b128 streaming.
// Each block emits one partial per loss term (deterministic two-pass reduce).
// ---------------------------------------------------------------------------
__global__ __launch_bounds__(256)
void fused_loss_kernel(const float* __restrict__ logits,
                       const int* __restrict__ gtc,
                       const float4* __restrict__ pb,
                       const float4* __restrict__ gb,
                       const float* __restrict__ cen,
                       float* __restrict__ partF,
                       float* __restrict__ partG,
                       float* __restrict__ partC,
                       int n) {
    __shared__ float smem[WAVES * 2 * ROWS * NCLS]; // 40 KB staging
    __shared__ float redf[WAVES], redg[WAVES], redc[WAVES];

    const int lane   = threadIdx.x & 31;
    const int wid    = threadIdx.x >> 5;
    const int gwave  = blockIdx.x * WAVES + wid;
    const int nwaves = gridDim.x * WAVES;
    const int ntiles = n / ROWS;

    // LDS byte address of this wave's staging region (low 32 bits of the
    // flat pointer to shared memory == LDS byte offset on gfx1250).
    const uint32_t ldsWave =
        (uint32_t)(uintptr_t)(&smem[wid * 2 * ROWS * NCLS]);

    float accF = 0.0f;

    auto issue = [&](int t, int b) {
        uint32_t laddr = ldsWave + (uint32_t)(b * TILE_BYTES) + (uint32_t)(lane * 16);
        uint32_t goff  = (uint32_t)t * (uint32_t)TILE_BYTES + (uint32_t)(lane * 16);
#pragma unroll
        for (int k = 0; k < COPIES; ++k) {
            // GVS mode: 64-bit SGPR base + 32-bit per-lane byte offset.
            // Each instruction copies 512B (wave32 x 16B) global -> LDS,
            // tracked on ASYNCcnt.
            asm volatile("global_load_async_to_lds_b128 %0, %1, %2"
                         :
                         : "v"(laddr + (uint32_t)(k * 512)),
                           "v"(goff + (uint32_t)(k * 512)),
                           "s"(logits)
                         : "memory");
        }
    };

    // ---- Part 1: focal loss over the logits stream -------------------------
    int buf = 0;
    if (gwave < ntiles) issue(gwave, 0);

    for (int t = gwave; t < ntiles; t += nwaves) {
        const int nxt = t + nwaves;
        if (nxt < ntiles) {
            issue(nxt, buf ^ 1);                              // prefetch next tile
            asm volatile("s_wait_asynccnt 5" ::: "memory");   // current tile landed
        } else {
            asm volatile("s_wait_asynccnt 0" ::: "memory");
        }

        const float* rowp = &smem[wid * 2 * ROWS * NCLS + buf * ROWS * NCLS];
#pragma unroll
        for (int r = 0; r < ROWS; ++r) {
            // lane l covers classes l, l+32, l+64(<80)
            float x0 = rowp[r * NCLS + lane];
            float x1 = rowp[r * NCLS + 32 + lane];
            float x2 = (lane < 16) ? rowp[r * NCLS + 64 + lane] : -__builtin_inff();

            float m = fmaxf(fmaxf(x0, x1), x2);
#pragma unroll
            for (int off = 16; off > 0; off >>= 1)
                m = fmaxf(m, __shfl_xor(m, off, 32));

            float s = __expf(x0 - m) + __expf(x1 - m) + __expf(x2 - m);
            s = warp_sum_f32(s);
            float lse = m + __logf(s);

            int gt = gtc[t * ROWS + r];                       // wave-uniform scalar load
            float selv = (gt < 32) ? x0 : ((gt < 64) ? x1 : x2);
            float xg = __shfl(selv, gt & 31, 32);

            float ce = lse - xg;
            float pt = __expf(-ce);
            float om = 1.0f - pt;
            if (lane == 0) accF += 0.25f * om * om * ce;      // ALPHA*(1-pt)^GAMMA*ce
        }
        asm volatile("s_wait_dscnt 0" ::: "memory");          // LDS reads done before reuse
        buf ^= 1;
    }

    // Tail rows (n % ROWS), thread-per-row serial fallback.
    const int rem  = n - ntiles * ROWS;
    const int gtid = blockIdx.x * blockDim.x + threadIdx.x;
    if (gtid < rem) {
        const int row = ntiles * ROWS + gtid;
        const float* rp = logits + (size_t)row * NCLS;
        float m = -__builtin_inff();
        for (int c = 0; c < NCLS; ++c) m = fmaxf(m, rp[c]);
        float s = 0.0f;
        for (int c = 0; c < NCLS; ++c) s += __expf(rp[c] - m);
        float lse = m + __logf(s);
        float ce  = lse - rp[gtc[row]];
        float pt  = __expf(-ce);
        float om  = 1.0f - pt;
        accF += 0.25f * om * om * ce;
    }

    // ---- Part 2: GIoU + centerness streaming ------------------------------
    float accG = 0.0f, accC = 0.0f;
    for (int i = gtid; i < n; i += gridDim.x * blockDim.x) {
        float4 p = pb[i];
        float4 g = gb[i];
        float iw    = fmaxf(fminf(p.z, g.z) - fmaxf(p.x, g.x), 0.0f);
        float ih    = fmaxf(fminf(p.w, g.w) - fmaxf(p.y, g.y), 0.0f);
        float inter = iw * ih;
        float ap    = (p.z - p.x) * (p.w - p.y);
        float ag    = (g.z - g.x) * (g.w - g.y);
        float uni   = ap + ag - inter;
        float iou   = inter / (uni + 1e-7f);
        float cw    = fmaxf(p.z, g.z) - fminf(p.x, g.x);
        float ch    = fmaxf(p.w, g.w) - fminf(p.y, g.y);
        float ac    = cw * ch;
        float giou  = iou - (ac - uni) / (ac + 1e-7f);
        accG += 1.0f - giou;

        float tt = -cen[i];                                    // logaddexp(0,-c), stable
        accC += fmaxf(tt, 0.0f) + log1pf(__expf(-fabsf(tt)));
    }

    // ---- Block reduction of all three partial sums ------------------------
    accF = warp_sum_f32(accF);
    accG = warp_sum_f32(accG);
    accC = warp_sum_f32(accC);
    if (lane == 0) { redf[wid] = accF; redg[wid] = accG; redc[wid] = accC; }
    __syncthreads();
    if (threadIdx.x == 0) {
        float sf = 0.0f, sg = 0.0f, sc = 0.0f;
        for (int i = 0; i < WAVES; ++i) { sf += redf[i]; sg += redg[i]; sc += redc[i]; }
        partF[blockIdx.x] = sf;
        partG[blockIdx.x] = sg;
        partC[blockIdx.x] = sc;
    }
}

// ---------------------------------------------------------------------------
// Final deterministic reduction of per-block partials -> 4 scalars
// ---------------------------------------------------------------------------
__global__ __launch_bounds__(256)
void finalize_kernel(const float* __restrict__ ws, float* __restrict__ out,
                     int nb, int n) {
    __shared__ double red0[WAVES], red1[WAVES], red2[WAVES];
    double f = 0.0, g = 0.0, c = 0.0;
    for (int i = threadIdx.x; i < nb; i += blockDim.x) f += (double)ws[i];
    for (int i = threadIdx.x; i < nb; i += blockDim.x) g += (double)ws[nb + i];
    for (int i = threadIdx.x; i < nb; i += blockDim.x) c += (double)ws[2 * nb + i];
#pragma unroll
    for (int off = 16; off > 0; off >>= 1) {
        f += __shfl_xor(f, off, 32);
        g += __shfl_xor(g, off, 32);
        c += __shfl_xor(c, off, 32);
    }
    const int lane = threadIdx.x & 31, wid = threadIdx.x >> 5;
    if (lane == 0) { red0[wid] = f; red1[wid] = g; red2[wid] = c; }
    __syncthreads();
    if (threadIdx.x == 0) {
        double sf = 0.0, sg = 0.0, sc = 0.0;
        for (int i = 0; i < WAVES; ++i) { sf += red0[i]; sg += red1[i]; sc += red2[i]; }
        const double inv = 1.0 / (double)n;
        const float fm = (float)(sf * inv);
        const float gm = (float)(sg * inv);
        const float cm = (float)(sc * inv);
        out[0] = fm + gm + 0.1f * cm;   // total
        out[1] = fm;                    // focal
        out[2] = gm;                    // iou
        out[3] = cm;                    // center
    }
}

extern "C" void kernel_launch(void* const* d_in, const int* in_sizes, int n_in,
                              void* d_out, int out_size, void* d_ws, size_t ws_size,
                              hipStream_t stream) {
    (void)n_in; (void)out_size; (void)ws_size;
    const float*  cls = (const float*)d_in[0];
    const float4* pb  = (const float4*)d_in[1];
    const float*  cen = (const float*)d_in[2];
    const float4* gb  = (const float4*)d_in[3];
    const int*    gtc = (const int*)d_in[4];
    const int n = in_sizes[2];            // centerness length == N

    const int NB = 1024;                  // fused-kernel blocks
    float* ws = (float*)d_ws;             // [0,NB) focal | [NB,2NB) giou | [2NB,3NB) center

    fused_loss_kernel<<<NB, 256, 0, stream>>>(cls, gtc, pb, gb, cen,
                                              ws, ws + NB, ws + 2 * NB, n);
    finalize_kernel<<<1, 256, 0, stream>>>(ws, (float*)d_out, NB, n);
}